// RotPointMatchingLoss_66614942760997
// MI455X (gfx1250) — compile-verified
//
#include <hip/hip_runtime.h>
#include <math.h>

// Problem constants from the reference.
#define B_DIM 16
#define N_DIM 4096
#define NTILE (N_DIM / 16)       // 256 16-wide tiles per batch
#define SWZ_PER_B (NTILE * 32)   // float2 slots per batch in swizzled layout

typedef __attribute__((ext_vector_type(2))) float v2f;
typedef __attribute__((ext_vector_type(8))) float v8f;

// ---------------------------------------------------------------------------
// Kernel 1: rotate points, emit WMMA-ready swizzled operand layouts, and
// fully precompute the per-point direct loss (loss_dir needs no KNN).
//
// 16x4 f32 A layout: lane L<16 holds (K0,K1)=(p.x,p.y) of row m=L;
//                    lane L+16 holds (K2,K3)=(p.z,|p|^2) of row m=L.
// 4x16 f32 B layout: lane L<16 holds (K0,K1)=(-2t.x,-2t.y) of col n=L;
//                    lane L+16 holds (K2,K3)=(-2t.z, 1)    of col n=L.
// With C seeded to |t_n|^2, D = A*B + C = exact squared distance d2[m,n].
// ---------------------------------------------------------------------------
__global__ __launch_bounds__(256) void rpml_prep(
    const float* __restrict__ pred_r, const float* __restrict__ gt_r,
    const float* __restrict__ points, float2* __restrict__ Aswz,
    float2* __restrict__ Bswz, float* __restrict__ tnorm2,
    float* __restrict__ dirDist) {
  int gid = blockIdx.x * blockDim.x + threadIdx.x;
  if (gid >= B_DIM * N_DIM) return;
  int b = gid >> 12;          // gid / 4096
  int n = gid & (N_DIM - 1);  // point index within batch
  const float* Rp = pred_r + b * 9;
  const float* Rg = gt_r + b * 9;
  float x0 = points[3 * gid + 0];
  float x1 = points[3 * gid + 1];
  float x2 = points[3 * gid + 2];
  float px = Rp[0] * x0 + Rp[1] * x1 + Rp[2] * x2;
  float py = Rp[3] * x0 + Rp[4] * x1 + Rp[5] * x2;
  float pz = Rp[6] * x0 + Rp[7] * x1 + Rp[8] * x2;
  float tx = Rg[0] * x0 + Rg[1] * x1 + Rg[2] * x2;
  float ty = Rg[3] * x0 + Rg[4] * x1 + Rg[5] * x2;
  float tz = Rg[6] * x0 + Rg[7] * x1 + Rg[8] * x2;

  int tile = n >> 4;
  int col = n & 15;
  int base = b * SWZ_PER_B + tile * 32 + col;
  Aswz[base] = make_float2(px, py);
  Aswz[base + 16] = make_float2(pz, px * px + py * py + pz * pz);
  Bswz[base] = make_float2(-2.0f * tx, -2.0f * ty);
  Bswz[base + 16] = make_float2(-2.0f * tz, 1.0f);
  tnorm2[gid] = tx * tx + ty * ty + tz * tz;

  float dx = px - tx, dy = py - ty, dz = pz - tz;
  dirDist[gid] = sqrtf(dx * dx + dy * dy + dz * dz);
}

// ---------------------------------------------------------------------------
// Kernel 2: one wave32 per (batch, pair of n-tiles). Each loaded pred A-tile
// feeds two V_WMMA_F32_16X16X4_F32 (one per target column tile); C carries
// |t_n|^2 so D is the true d2 tile. No argmin needed: the matched distance
// in the reference equals sqrt(min_m d2) -- a pure fmin tree per tile.
// EXEC stays all-ones throughout (wave32 WMMA requirement).
// ---------------------------------------------------------------------------
__global__ __launch_bounds__(256) void rpml_knn(
    const float2* __restrict__ Aswz, const float2* __restrict__ Bswz,
    const float* __restrict__ tnorm2, const float* __restrict__ dirDist,
    const float* __restrict__ mesh_diam, const int* __restrict__ labels,
    const int* __restrict__ sym_flags, float* __restrict__ partials) {
  int wave = (blockIdx.x * blockDim.x + threadIdx.x) >> 5;  // 0..2047
  int lane = threadIdx.x & 31;
  int b = wave >> 7;         // wave / 128
  int wpair = wave & 127;    // handles n-tiles 2*wpair, 2*wpair+1
  int col = lane & 15;
  int hi = lane >> 4;

  const float2* As = Aswz + b * SWZ_PER_B;
  const float2* Bs = Bswz + b * SWZ_PER_B;
  int n0 = wpair * 32 + col;  // column of tile 2*wpair
  int n1 = n0 + 16;           // column of tile 2*wpair+1

  // B operands (fixed per wave), already in WMMA lane layout.
  float2 b0 = Bs[(2 * wpair) * 32 + lane];
  float2 b1 = Bs[(2 * wpair + 1) * 32 + lane];
  v2f Bv0, Bv1;
  Bv0[0] = b0.x; Bv0[1] = b0.y;
  Bv1[0] = b1.x; Bv1[1] = b1.y;

  // C = |t_n|^2 broadcast down each column -> WMMA emits true d2.
  float tn0 = tnorm2[b * N_DIM + n0];
  float tn1 = tnorm2[b * N_DIM + n1];
  v8f c0, c1;
#pragma unroll
  for (int i = 0; i < 8; ++i) { c0[i] = tn0; c1[i] = tn1; }

  float best0 = 3.402823466e+38f;
  float best1 = 3.402823466e+38f;

  for (int mt = 0; mt < NTILE; ++mt) {
    float2 av = As[mt * 32 + lane];
    v2f Av;
    Av[0] = av.x;
    Av[1] = av.y;
    // (neg_a, A, neg_b, B, c_mod, C, reuse_a, reuse_b)
    v8f d0 = __builtin_amdgcn_wmma_f32_16x16x4_f32(false, Av, false, Bv0,
                                                   (short)0, c0, false, false);
    v8f d1 = __builtin_amdgcn_wmma_f32_16x16x4_f32(false, Av, false, Bv1,
                                                   (short)0, c1, false, false);
    float m0 = fminf(fminf(fminf(d0[0], d0[1]), fminf(d0[2], d0[3])),
                     fminf(fminf(d0[4], d0[5]), fminf(d0[6], d0[7])));
    float m1 = fminf(fminf(fminf(d1[0], d1[1]), fminf(d1[2], d1[3])),
                     fminf(fminf(d1[4], d1[5]), fminf(d1[6], d1[7])));
    best0 = fminf(best0, m0);
    best1 = fminf(best1, m1);
  }

  // Merge the two half-wave row groups (rows 0..7 vs 8..15 of each tile).
  best0 = fminf(best0, __shfl_xor(best0, 16, 32));
  best1 = fminf(best1, __shfl_xor(best1, 16, 32));

  float dknn0 = sqrtf(fmaxf(best0, 0.0f));  // clamp fp cancellation
  float dknn1 = sqrtf(fmaxf(best1, 0.0f));
  float ddir0 = dirDist[b * N_DIM + n0];
  float ddir1 = dirDist[b * N_DIM + n1];

  int lbl = labels[b];
  bool sym = (sym_flags[lbl] != 0);
  float inv = 1.0f / (mesh_diam[lbl] * (float)N_DIM);
  float contrib = ((sym ? dknn0 : ddir0) + (sym ? dknn1 : ddir1)) * inv;
  if (hi) contrib = 0.0f;  // lanes 16..31 duplicate the same columns

  // Wave reduction (fixed order -> deterministic).
  contrib += __shfl_xor(contrib, 1, 32);
  contrib += __shfl_xor(contrib, 2, 32);
  contrib += __shfl_xor(contrib, 4, 32);
  contrib += __shfl_xor(contrib, 8, 32);
  contrib += __shfl_xor(contrib, 16, 32);
  if (lane == 0) partials[wave] = contrib;
}

// ---------------------------------------------------------------------------
// Kernel 3: deterministic tree reduction of 2048 per-wave partials -> scalar.
// ---------------------------------------------------------------------------
__global__ __launch_bounds__(256) void rpml_reduce(
    const float* __restrict__ partials, float* __restrict__ out) {
  __shared__ float sdata[256];
  int tid = threadIdx.x;
  float s = 0.0f;
  for (int i = tid; i < B_DIM * (NTILE / 2); i += 256) s += partials[i];
  sdata[tid] = s;
  __syncthreads();
  for (int off = 128; off > 0; off >>= 1) {
    if (tid < off) sdata[tid] += sdata[tid + off];
    __syncthreads();
  }
  if (tid == 0) out[0] = sdata[0] / (float)B_DIM;
}

extern "C" void kernel_launch(void* const* d_in, const int* in_sizes, int n_in,
                              void* d_out, int out_size, void* d_ws,
                              size_t ws_size, hipStream_t stream) {
  const float* pred_r = (const float*)d_in[0];     // [16,3,3]
  const float* gt_r = (const float*)d_in[1];       // [16,3,3]
  const float* points = (const float*)d_in[2];     // [16,4096,3]
  const float* mesh_diam = (const float*)d_in[3];  // [21]
  const int* labels = (const int*)d_in[4];         // [16]
  const int* sym_flags = (const int*)d_in[5];      // [21] (0/1)
  float* out = (float*)d_out;

  // Workspace carve-up (~2.5 MB): Aswz | Bswz | tnorm2 | dirDist | partials
  char* ws = (char*)d_ws;
  size_t swz_bytes = (size_t)B_DIM * SWZ_PER_B * sizeof(float2);  // 1 MB each
  size_t vec_bytes = (size_t)B_DIM * N_DIM * sizeof(float);       // 256 KB each
  float2* Aswz = (float2*)(ws);
  float2* Bswz = (float2*)(ws + swz_bytes);
  float* tnorm2 = (float*)(ws + 2 * swz_bytes);
  float* dirDist = (float*)(ws + 2 * swz_bytes + vec_bytes);
  float* partials = (float*)(ws + 2 * swz_bytes + 2 * vec_bytes);

  // 1) rotate + swizzle + direct loss: 65536 points
  rpml_prep<<<(B_DIM * N_DIM) / 256, 256, 0, stream>>>(
      pred_r, gt_r, points, Aswz, Bswz, tnorm2, dirDist);

  // 2) WMMA KNN min-distance: 2048 waves = 256 blocks x 8 wave32
  rpml_knn<<<(B_DIM * (NTILE / 2) * 32) / 256, 256, 0, stream>>>(
      Aswz, Bswz, tnorm2, dirDist, mesh_diam, labels, sym_flags, partials);

  // 3) final reduction to scalar
  rpml_reduce<<<1, 256, 0, stream>>>(partials, out);
}